// vPFAE_EC_68539088110354
// MI455X (gfx1250) — compile-verified
//
#include <hip/hip_runtime.h>

typedef __attribute__((ext_vector_type(2))) float v2f;
typedef __attribute__((ext_vector_type(8))) float v8f;

#define NNODES 8192
#define NEDGES 65536
#define INC    64
#define OUTC   16
#define EFEAT  8
#define L1C    48
#define L2C    32

// Packed GEMM B widths: [EF*Cout | nn_b proj (Cout) | root (Cout)]
#define W1COLS (EFEAT*L1C + L1C + L1C)   // 480
#define W2COLS (EFEAT*L2C + L2C + L2C)   // 320
#define WHCOLS 32                         // mu (16) | ls (16)

// ---------------- workspace layout (float offsets) ----------------
#define OFF_WT1  0u
#define OFF_WT2  (OFF_WT1 + INC*W1COLS)            // 30720
#define OFF_WH   (OFF_WT2 + L1C*W2COLS)            // 46080
#define OFF_Y1   (OFF_WH  + L2C*WHCOLS)            // 47104
#define OFF_Y2   (OFF_Y1  + (size_t)NNODES*W1COLS) // + 3932160
#define OFF_AGG1 (OFF_Y2  + (size_t)NNODES*W2COLS) // + 2621440
#define OFF_H1   (OFF_AGG1 + (size_t)NNODES*L1C)
#define OFF_AGG2 (OFF_H1   + (size_t)NNODES*L1C)
#define OFF_H2   (OFF_AGG2 + (size_t)NNODES*L2C)
#define OFF_HH   (OFF_H2   + (size_t)NNODES*L2C)
#define OFF_AGGG (OFF_HH   + (size_t)NNODES*L2C)
#define OFF_DEG  (OFF_AGGG + (size_t)NNODES*L2C)
#define OFF_DINV (OFF_DEG  + (size_t)NNODES)

// ---------------- weight repack kernels ----------------
__global__ void repack1(const float* __restrict__ nn1_w, const float* __restrict__ nn1_b,
                        const float* __restrict__ root1, float* __restrict__ Wt1) {
  int idx = blockIdx.x * blockDim.x + threadIdx.x;
  if (idx >= INC * W1COLS) return;
  int i = idx / W1COLS, c = idx % W1COLS;
  float v;
  if (c < EFEAT * L1C)            { int f = c / L1C, o = c % L1C; v = nn1_w[f * (INC * L1C) + i * L1C + o]; }
  else if (c < EFEAT * L1C + L1C) { int o = c - EFEAT * L1C;      v = nn1_b[i * L1C + o]; }
  else                            { int o = c - EFEAT * L1C - L1C; v = root1[i * L1C + o]; }
  Wt1[idx] = v;
}

__global__ void repack2(const float* __restrict__ nn2_w, const float* __restrict__ nn2_b,
                        const float* __restrict__ root2, float* __restrict__ Wt2) {
  int idx = blockIdx.x * blockDim.x + threadIdx.x;
  if (idx >= L1C * W2COLS) return;
  int i = idx / W2COLS, c = idx % W2COLS;
  float v;
  if (c < EFEAT * L2C)            { int f = c / L2C, o = c % L2C; v = nn2_w[f * (L1C * L2C) + i * L2C + o]; }
  else if (c < EFEAT * L2C + L2C) { int o = c - EFEAT * L2C;      v = nn2_b[i * L2C + o]; }
  else                            { int o = c - EFEAT * L2C - L2C; v = root2[i * L2C + o]; }
  Wt2[idx] = v;
}

__global__ void repackH(const float* __restrict__ mu_w, const float* __restrict__ ls_w,
                        float* __restrict__ Wh) {
  int idx = blockIdx.x * blockDim.x + threadIdx.x;
  if (idx >= L2C * WHCOLS) return;
  int i = idx / WHCOLS, c = idx % WHCOLS;
  Wh[idx] = (c < OUTC) ? mu_w[i * OUTC + c] : ls_w[i * OUTC + (c - OUTC)];
}

// ---------------- fp32 WMMA GEMM: C[M,ncols] = A[M,K] @ B[K,ncols] ----------------
// 4 waves per block; each wave owns one 16x16 tile; V_WMMA_F32_16X16X4_F32 over K.
// A fragment (16x4 fp32): lanes 0-15 hold K=0,1; lanes 16-31 hold K=2,3.
// C/D fragment: VGPR r, lane l -> M = r + 8*(l>=16), N = l%16.
__global__ void wmma_gemm_f32(const float* __restrict__ A, const float* __restrict__ B,
                              float* __restrict__ C, int K, int ncols, int tilesN) {
  int wave = threadIdx.x >> 5;
  int lane = threadIdx.x & 31;
  int tnIdx = blockIdx.y * 4 + wave;
  if (tnIdx >= tilesN) return;              // wave-uniform: EXEC stays all-1s
  int tm = blockIdx.x << 4;
  int tn = tnIdx << 4;
  int mr = lane & 15;
  int kh = (lane >> 4) << 1;                // 0 or 2
  const float* arow = A + (size_t)(tm + mr) * K;
  v8f acc = {};
  for (int k0 = 0; k0 < K; k0 += 4) {
    int ka = k0 + kh;
    v2f a;
    a.x = arow[ka];
    a.y = arow[ka + 1];
    v2f b;
    b.x = B[(size_t)ka * ncols + tn + mr];
    b.y = B[(size_t)(ka + 1) * ncols + tn + mr];
    acc = __builtin_amdgcn_wmma_f32_16x16x4_f32(false, a, false, b, (short)0, acc, false, false);
  }
  int mb = tm + ((lane >> 4) << 3);
  int nc = tn + mr;
#pragma unroll
  for (int r = 0; r < 8; ++r)
    C[(size_t)(mb + r) * ncols + nc] = acc[r];
}

// ---------------- layer-1 edge scatter (+ fused degree count) ----------------
// block(64,4): lanes 0..47 = output channel, lane 48 counts in-degree.
__global__ void edge_pass1(const int* __restrict__ src, const int* __restrict__ dst,
                           const float* __restrict__ ea, const float* __restrict__ Y1,
                           float* __restrict__ agg1, float* __restrict__ deg) {
  int e = blockIdx.x * blockDim.y + threadIdx.y;
  if (e >= NEDGES) return;
  int o = threadIdx.x;
  int s = src[e], d = dst[e];
  if (o < L1C) {
    const float* y = Y1 + (size_t)s * W1COLS;
    float acc = y[EFEAT * L1C + o];         // per-edge nn_b projection term
#pragma unroll
    for (int f = 0; f < EFEAT; ++f)
      acc += ea[e * EFEAT + f] * y[f * L1C + o];
    atomicAdd(&agg1[d * L1C + o], acc);
  } else if (o == L1C) {
    atomicAdd(&deg[d], 1.0f);
  }
}

__global__ void node1(const float* __restrict__ agg1, const float* __restrict__ Y1,
                      const float* __restrict__ bias1, float* __restrict__ h1) {
  int idx = blockIdx.x * blockDim.x + threadIdx.x;
  if (idx >= NNODES * L1C) return;
  int n = idx / L1C, o = idx % L1C;
  float v = agg1[idx] + Y1[(size_t)n * W1COLS + (EFEAT * L1C + L1C) + o] + bias1[o];
  h1[idx] = v > 0.0f ? v : 0.0f;
}

// ---------------- layer-2 edge scatter: block(32,8), lane = channel ----------------
__global__ void edge_pass2(const int* __restrict__ src, const int* __restrict__ dst,
                           const float* __restrict__ ea, const float* __restrict__ Y2,
                           float* __restrict__ agg2) {
  int e = blockIdx.x * blockDim.y + threadIdx.y;
  if (e >= NEDGES) return;
  int o = threadIdx.x;
  int s = src[e], d = dst[e];
  const float* y = Y2 + (size_t)s * W2COLS;
  float acc = y[EFEAT * L2C + o];
#pragma unroll
  for (int f = 0; f < EFEAT; ++f)
    acc += ea[e * EFEAT + f] * y[f * L2C + o];
  atomicAdd(&agg2[d * L2C + o], acc);
}

__global__ void node2(const float* __restrict__ agg2, const float* __restrict__ Y2,
                      const float* __restrict__ bias2, const float* __restrict__ deg,
                      float* __restrict__ h2, float* __restrict__ dinv) {
  int idx = blockIdx.x * blockDim.x + threadIdx.x;
  if (idx >= NNODES * L2C) return;
  int n = idx >> 5, o = idx & 31;
  float v = agg2[idx] + Y2[(size_t)n * W2COLS + (EFEAT * L2C + L2C) + o] + bias2[o];
  h2[idx] = v > 0.0f ? v : 0.0f;
  if (o == 0) dinv[n] = rsqrtf(deg[n] + 1.0f);
}

// ---------------- GCN edge scatter (both heads packed, 32 channels) ----------------
__global__ void gcn_edge(const int* __restrict__ src, const int* __restrict__ dst,
                         const float* __restrict__ hh, const float* __restrict__ dinv,
                         float* __restrict__ aggG) {
  int e = blockIdx.x * blockDim.y + threadIdx.y;
  if (e >= NEDGES) return;
  int c = threadIdx.x;
  int s = src[e], d = dst[e];
  float w = dinv[s] * dinv[d];
  atomicAdd(&aggG[d * 32 + c], hh[s * 32 + c] * w);
}

__global__ void final_write(const float* __restrict__ aggG, const float* __restrict__ hh,
                            const float* __restrict__ deg, const float* __restrict__ mu_b,
                            const float* __restrict__ ls_b, float* __restrict__ out) {
  int idx = blockIdx.x * blockDim.x + threadIdx.x;
  if (idx >= NNODES * 32) return;
  int n = idx >> 5, c = idx & 31;
  float v = aggG[idx] + hh[idx] / (deg[n] + 1.0f);
  if (c < OUTC) out[n * OUTC + c] = v + mu_b[c];
  else          out[NNODES * OUTC + n * OUTC + (c - OUTC)] = v + ls_b[c - OUTC];
}

// ---------------- launch ----------------
extern "C" void kernel_launch(void* const* d_in, const int* in_sizes, int n_in,
                              void* d_out, int out_size, void* d_ws, size_t ws_size,
                              hipStream_t stream) {
  const float* x     = (const float*)d_in[0];
  const int*   ei    = (const int*)  d_in[1];
  const float* ea    = (const float*)d_in[2];
  const float* nn1_w = (const float*)d_in[3];
  const float* nn1_b = (const float*)d_in[4];
  const float* root1 = (const float*)d_in[5];
  const float* bias1 = (const float*)d_in[6];
  const float* nn2_w = (const float*)d_in[7];
  const float* nn2_b = (const float*)d_in[8];
  const float* root2 = (const float*)d_in[9];
  const float* bias2 = (const float*)d_in[10];
  const float* mu_w  = (const float*)d_in[11];
  const float* mu_b  = (const float*)d_in[12];
  const float* ls_w  = (const float*)d_in[13];
  const float* ls_b  = (const float*)d_in[14];
  const int* src = ei;
  const int* dst = ei + NEDGES;

  float* ws   = (float*)d_ws;
  float* Wt1  = ws + OFF_WT1;
  float* Wt2  = ws + OFF_WT2;
  float* Wh   = ws + OFF_WH;
  float* Y1   = ws + OFF_Y1;
  float* Y2   = ws + OFF_Y2;
  float* agg1 = ws + OFF_AGG1;
  float* h1   = ws + OFF_H1;
  float* agg2 = ws + OFF_AGG2;
  float* h2   = ws + OFF_H2;
  float* hh   = ws + OFF_HH;
  float* aggG = ws + OFF_AGGG;
  float* deg  = ws + OFF_DEG;
  float* dinv = ws + OFF_DINV;

  // zero the scatter accumulators each call (graph-capture safe)
  hipMemsetAsync(agg1, 0, sizeof(float) * (size_t)NNODES * L1C, stream);
  hipMemsetAsync(agg2, 0, sizeof(float) * (size_t)NNODES * L2C, stream);
  hipMemsetAsync(aggG, 0, sizeof(float) * (size_t)NNODES * L2C, stream);
  hipMemsetAsync(deg,  0, sizeof(float) * (size_t)NNODES, stream);

  // weight repacks (tiny)
  repack1<<<(INC * W1COLS + 255) / 256, 256, 0, stream>>>(nn1_w, nn1_b, root1, Wt1);
  repack2<<<(L1C * W2COLS + 255) / 256, 256, 0, stream>>>(nn2_w, nn2_b, root2, Wt2);
  repackH<<<(L2C * WHCOLS + 255) / 256, 256, 0, stream>>>(mu_w, ls_w, Wh);

  // Layer 1: Y1 = x @ Wt1   [8192,64]x[64,480]
  {
    int tilesN = W1COLS / 16;   // 30
    dim3 grid(NNODES / 16, (tilesN + 3) / 4);
    wmma_gemm_f32<<<grid, 128, 0, stream>>>(x, Wt1, Y1, INC, W1COLS, tilesN);
  }
  edge_pass1<<<NEDGES / 4, dim3(64, 4), 0, stream>>>(src, dst, ea, Y1, agg1, deg);
  node1<<<(NNODES * L1C + 255) / 256, 256, 0, stream>>>(agg1, Y1, bias1, h1);

  // Layer 2: Y2 = h1 @ Wt2  [8192,48]x[48,320]
  {
    int tilesN = W2COLS / 16;   // 20
    dim3 grid(NNODES / 16, (tilesN + 3) / 4);
    wmma_gemm_f32<<<grid, 128, 0, stream>>>(h1, Wt2, Y2, L1C, W2COLS, tilesN);
  }
  edge_pass2<<<NEDGES / 8, dim3(32, 8), 0, stream>>>(src, dst, ea, Y2, agg2);
  node2<<<(NNODES * L2C + 255) / 256, 256, 0, stream>>>(agg2, Y2, bias2, deg, h2, dinv);

  // Heads: hh = h2 @ Wh     [8192,32]x[32,32]
  {
    int tilesN = WHCOLS / 16;   // 2
    dim3 grid(NNODES / 16, 1);
    wmma_gemm_f32<<<grid, 128, 0, stream>>>(h2, Wh, hh, L2C, WHCOLS, tilesN);
  }
  gcn_edge<<<NEDGES / 8, dim3(32, 8), 0, stream>>>(src, dst, hh, dinv, aggG);
  final_write<<<(NNODES * 32 + 255) / 256, 256, 0, stream>>>(aggG, hh, deg, mu_b, ls_b,
                                                             (float*)d_out);
}